// FeatureSpaceEntropy_4166118277841
// MI455X (gfx1250) — compile-verified
//
#include <hip/hip_runtime.h>
#include <hip/hip_bf16.h>

typedef __attribute__((ext_vector_type(2))) float v2f;
typedef __attribute__((ext_vector_type(8))) float v8f;

static constexpr int kHW    = 512 * 1024; // full map pixels
static constexpr int kC     = 512;        // feature dim
static constexpr int kCls   = 19;         // classes
static constexpr int kSamp  = 8192;       // sampled pixels
static constexpr int kBStr  = 516;        // LDS proto row stride (pad: 4n+k bank spread)

// ---------------------------------------------------------------- zero output
__global__ void fse_zero_kernel(float* __restrict__ out, int n) {
  int i = blockIdx.x * blockDim.x + threadIdx.x;
  if (i < n) out[i] = 0.0f;
}

// ------------------------------------------- per-sample argmax + proto norms
__global__ void fse_prep_kernel(const float* __restrict__ outputs,
                                const float* __restrict__ protos,
                                const int* __restrict__ sidx,
                                int* __restrict__ cid,
                                float* __restrict__ protosq) {
  int i = blockIdx.x * blockDim.x + threadIdx.x;
  if (i < kSamp) {
    int p = sidx[i];
    float best = outputs[p];
    int bk = 0;
#pragma unroll
    for (int k = 1; k < kCls; ++k) {
      float v = outputs[(size_t)k * kHW + p];
      if (v > best) { best = v; bk = k; }   // first-max semantics like argmax
    }
    cid[i] = bk;
  } else if (i < kSamp + kCls) {
    int k = i - kSamp;
    float s = 0.0f;
    for (int c = 0; c < kC; ++c) {
      float v = protos[k * kC + c];
      s += v * v;
    }
    protosq[k] = s;
  }
}

// ----------------------------- WMMA distance GEMM + rank + scatter, per block:
// 128 samples, 8 waves, each wave does a 16(M) x 32(N, 19 valid) x 512(K) tile
// with V_WMMA_F32_16X16X4_F32.
__global__ void __launch_bounds__(256)
fse_wmma_rank_kernel(const float* __restrict__ features,
                     const float* __restrict__ protos,
                     const int* __restrict__ sidx,
                     const int* __restrict__ cid,
                     const float* __restrict__ protosq,
                     float* __restrict__ out) {
  __shared__ float Bl[kCls][kBStr];   // Bl[n][k] = protos[n][k] (transposed stage)
  __shared__ float Dt[8][16][33];     // per-wave 16x32 dot-product tile (+pad)
  __shared__ float FsqP[8][32];       // per-lane partial |feat|^2

  const int tid = threadIdx.x;

  // Stage prototypes into LDS (coalesced over k).
  for (int i = tid; i < kCls * kC; i += 256) {
    int n = i >> 9;          // /512
    int k = i & (kC - 1);    // %512
    Bl[n][k] = protos[i];
  }
  __syncthreads();

  const int wave  = tid >> 5;
  const int lane  = tid & 31;
  const int laneM = lane & 15;
  const int isHi  = lane >> 4;          // 0: K=k0,k0+1  1: K=k0+2,k0+3

  const int m = blockIdx.x * 128 + wave * 16 + laneM;   // sample row (M)
  const int p = sidx[m];                                // pixel index

  // N-tile 1 covers classes 16..31; mask classes >= 19 to zero (VALU select,
  // no EXEC divergence -- WMMA requires EXEC all-ones).
  const int   n1    = 16 + laneM;
  const float mask1 = (n1 < kCls) ? 1.0f : 0.0f;
  const int   n1c   = (n1 < kCls) ? n1 : (kCls - 1);

  v8f c0 = {};     // classes 0..15
  v8f c1 = {};     // classes 16..31 (19.. are zero-padded)
  float fsq = 0.0f;
  const int kOff = isHi * 2;

  for (int k0 = 0; k0 < kC; k0 += 4) {
    const int kA = k0 + kOff;
    // strided gather: one cacheline per element; prefetch 16 planes ahead
    float ax = features[(size_t)kA * kHW + p];
    float ay = features[(size_t)(kA + 1) * kHW + p];
    int kpf = kA + 16; if (kpf > kC - 2) kpf = kC - 2;
    __builtin_prefetch(&features[(size_t)kpf * kHW + p], 0, 0);

    fsq += ax * ax + ay * ay;

    v2f a;  a.x = ax;                  a.y = ay;
    v2f b0; b0.x = Bl[laneM][kA];      b0.y = Bl[laneM][kA + 1];
    v2f b1; b1.x = Bl[n1c][kA] * mask1; b1.y = Bl[n1c][kA + 1] * mask1;

    c0 = __builtin_amdgcn_wmma_f32_16x16x4_f32(false, a, false, b0,
                                               (short)0, c0, false, false);
    c1 = __builtin_amdgcn_wmma_f32_16x16x4_f32(false, a, false, b1,
                                               (short)0, c1, false, false);
  }

  FsqP[wave][lane] = fsq;

  // C/D layout: VGPR e holds M = e + 8*isHi, N = lane % 16 (per N-tile).
#pragma unroll
  for (int e = 0; e < 8; ++e) {
    Dt[wave][e + 8 * isHi][laneM]      = c0[e];
    Dt[wave][e + 8 * isHi][16 + laneM] = c1[e];
  }
  __syncthreads();

  // Rank phase: one thread per sample (128 of 256 threads).
  if (tid < 128) {
    const int w    = tid >> 4;
    const int mloc = tid & 15;
    const int mg   = blockIdx.x * 128 + tid;
    const int myc  = cid[mg];
    const float fs = FsqP[w][mloc] + FsqP[w][mloc + 16];

    float d[kCls];
#pragma unroll
    for (int k = 0; k < kCls; ++k) {
      float v = fs - 2.0f * Dt[w][mloc][k] + protosq[k];
      d[k] = fmaxf(v, 1e-12f);          // clip(1e-12, None)
    }
    const float dc = d[myc];
    int rank = 0;
#pragma unroll
    for (int k = 0; k < kCls; ++k)
      rank += (d[k] < dc) || (d[k] == dc && k < myc);   // stable-sort position

    out[sidx[mg]] = (float)rank * (1.0f / (float)(kCls - 1));
  }
}

extern "C" void kernel_launch(void* const* d_in, const int* in_sizes, int n_in,
                              void* d_out, int out_size, void* d_ws, size_t ws_size,
                              hipStream_t stream) {
  const float* features = (const float*)d_in[0];   // (1,512,512,1024) f32
  const float* outputs  = (const float*)d_in[1];   // (1,19,512,1024)  f32
  const float* protos   = (const float*)d_in[2];   // (1,19,512)       f32
  const int*   sidx     = (const int*)d_in[3];     // (8192,)          i32
  float* out = (float*)d_out;                      // (512,1024)       f32

  int*   cid     = (int*)d_ws;
  float* protosq = (float*)((char*)d_ws + kSamp * sizeof(int));

  fse_zero_kernel<<<(kHW + 255) / 256, 256, 0, stream>>>(out, kHW);
  fse_prep_kernel<<<(kSamp + kCls + 255) / 256, 256, 0, stream>>>(
      outputs, protos, sidx, cid, protosq);
  fse_wmma_rank_kernel<<<kSamp / 128, 256, 0, stream>>>(
      features, protos, sidx, cid, protosq, out);
}